// S4Layer_18262200942823
// MI455X (gfx1250) — compile-verified
//
#include <hip/hip_runtime.h>
#include <math.h>

// Problem constants (match reference)
#define BB 8
#define LL 4096
#define HH 512
#define NN 32          // complex modes
#define T  64          // chunk length
#define CC 64          // chunks per sequence = LL/T
#define JJ 512         // BB*CC (GEMM1 "row" dim)
#define PP 32768       // BB*LL (positions)
#define KDIM 128       // 64 conv taps + 64 state reals

typedef float v2f __attribute__((ext_vector_type(2)));
typedef float v8f __attribute__((ext_vector_type(8)));

__device__ __forceinline__ v2f ld2(const float* p) {
  return *reinterpret_cast<const v2f*>(p);   // 8B-aligned b64 load
}
__device__ __forceinline__ float fast_sigmoid(float z) {
  return 1.0f / (1.0f + __expf(-z));         // v_exp_f32, branch-free
}
__device__ __forceinline__ float fast_gelu(float y) {
  // 0.5*y*(1+tanh(u)) == y*sigmoid(2u)
  const float u = 0.7978845608028654f * (y + 0.044715f * y * y * y);
  return y * fast_sigmoid(2.0f * u);
}

// ---------------------------------------------------------------------------
// Kernel 1: per-h coefficient table (tiny, precise libm).
//   G[h] is (128 x 64), k-major, t-contiguous:
//     k in [0,64):  G[k][t] = K[t-k] (t>=k) + D[h]*(k==t)   (Toeplitz + skip)
//     k = 64+2n:    G[k][t] =  2*Re(Ccoef_n * w_n^{t+1})
//     k = 64+2n+1:  G[k][t] = -2*Im(Ccoef_n * w_n^{t+1})
//   Wtab[h][n] = (w_re, w_im) for the scan kernel.
// ---------------------------------------------------------------------------
__global__ void s4_table_kernel(const float* __restrict__ log_dt,
                                const float* __restrict__ log_A_re,
                                const float* __restrict__ A_im,
                                const float* __restrict__ C_re,
                                const float* __restrict__ C_im,
                                const float* __restrict__ Dp,
                                float* __restrict__ G,
                                float* __restrict__ Wtab) {
  const int h = blockIdx.x;
  const int t = threadIdx.x;  // 0..63
  __shared__ float s_dtar[NN], s_dtai[NN], s_ccr[NN], s_cci[NN], s_K[T];

  if (t < NN) {
    const int n = t;
    const float dt   = expf(log_dt[h]);
    const float Are  = -expf(log_A_re[h * NN + n]);
    const float Aim  = A_im[h * NN + n];
    const float dtar = dt * Are, dtai = dt * Aim;
    const float er   = expf(dtar);
    const float wre  = er * cosf(dtai);
    const float wim  = er * sinf(dtai);
    // Ccoef = (C_re + i C_im) * (w - 1) / A
    const float dre = wre - 1.0f, dimg = wim;
    const float den = Are * Are + Aim * Aim;
    const float t1  = (dre * Are + dimg * Aim) / den;
    const float t2  = (dimg * Are - dre * Aim) / den;
    const float cr  = C_re[h * NN + n], ci = C_im[h * NN + n];
    s_ccr[n]  = cr * t1 - ci * t2;
    s_cci[n]  = cr * t2 + ci * t1;
    s_dtar[n] = dtar;
    s_dtai[n] = dtai;
    Wtab[(h * NN + n) * 2 + 0] = wre;
    Wtab[(h * NN + n) * 2 + 1] = wim;
  }
  __syncthreads();

  // K[t] = 2 * sum_n Re(Ccoef_n * exp(t * dtA_n))
  {
    float acc = 0.0f;
    for (int n = 0; n < NN; ++n) {
      const float er = expf((float)t * s_dtar[n]);
      const float pr = er * cosf((float)t * s_dtai[n]);
      const float pi = er * sinf((float)t * s_dtai[n]);
      acc += s_ccr[n] * pr - s_cci[n] * pi;
    }
    s_K[t] = 2.0f * acc;
  }
  __syncthreads();

  float* gB = G + h * (KDIM * T);
  const float dval = Dp[h];
  for (int k = 0; k < T; ++k) {
    float v = (t >= k) ? s_K[t - k] : 0.0f;
    if (k == t) v += dval;                 // fold skip connection into diagonal
    gB[k * T + t] = v;
  }
  for (int n = 0; n < NN; ++n) {
    const float er = expf((float)(t + 1) * s_dtar[n]);
    const float pr = er * cosf((float)(t + 1) * s_dtai[n]);
    const float pi = er * sinf((float)(t + 1) * s_dtai[n]);
    gB[(T + 2 * n)     * T + t] =  2.0f * (s_ccr[n] * pr - s_cci[n] * pi);
    gB[(T + 2 * n + 1) * T + t] = -2.0f * (s_ccr[n] * pi + s_cci[n] * pr);
  }
}

// ---------------------------------------------------------------------------
// Kernel 2: transpose x (B,L,H) -> Xt (H,64,512): Xt[h][u][b*64+c] = x[b][c*64+u][h]
// ---------------------------------------------------------------------------
__global__ void s4_xt_kernel(const float* __restrict__ x, float* __restrict__ Xt) {
  const int h0 = blockIdx.x * 32;
  const int c0 = blockIdx.y * 32;
  const int b  = blockIdx.z >> 6;
  const int u  = blockIdx.z & 63;
  __shared__ float tile[32][33];
  const int tx = threadIdx.x, ty = threadIdx.y;
#pragma unroll
  for (int i = 0; i < 4; ++i) {
    const int cl = ty + i * 8;
    tile[cl][tx] = x[b * (LL * HH) + ((c0 + cl) * T + u) * HH + h0 + tx];
  }
  __syncthreads();
#pragma unroll
  for (int i = 0; i < 4; ++i) {
    const int hl = ty + i * 8;
    Xt[(h0 + hl) * (T * JJ) + u * JJ + b * CC + c0 + tx] = tile[tx][hl];
  }
}

// ---------------------------------------------------------------------------
// Kernel 3: chunk-state scan. One wave32 per (b,h); lane n = complex mode n.
// ---------------------------------------------------------------------------
__global__ void s4_scan_kernel(const float* __restrict__ x,
                               const float* __restrict__ Wtab,
                               float* __restrict__ St) {
  const int wid = blockIdx.x * (blockDim.x >> 5) + (threadIdx.x >> 5);
  const int n   = threadIdx.x & 31;
  const int b   = wid >> 9;        // wid = b*512 + h
  const int h   = wid & 511;
  const float wre = Wtab[(h * NN + n) * 2 + 0];
  const float wim = Wtab[(h * NN + n) * 2 + 1];
  float* st = St + h * (T * JJ) + b * CC;
  st[(2 * n)     * JJ] = 0.0f;     // state entering chunk 0
  st[(2 * n + 1) * JJ] = 0.0f;
  float sr = 0.0f, si = 0.0f;
  const float* xb = x + b * (LL * HH) + h;
  for (int c = 0; c < CC; ++c) {
    for (int u = 0; u < T; ++u) {
      const float xv = xb[(c * T + u) * HH];   // wave-uniform (scalar) load
      const float nr = fmaf(wre, sr, fmaf(-wim, si, xv));
      const float ni = fmaf(wre, si, wim * sr);
      sr = nr; si = ni;
    }
    if (c < CC - 1) {
      st[(2 * n)     * JJ + c + 1] = sr;
      st[(2 * n + 1) * JJ + c + 1] = si;
    }
  }
}

// ---------------------------------------------------------------------------
// Kernel 4: per-h chunk GEMM + GELU. Wave computes a 64(j) x 64(t) tile:
// 16 v8f accumulators, per k-step 4 A-frags + 4 B-frags -> 16 WMMAs.
// Waves in a block share h and B-operand rows (consecutive j-tiles).
// ---------------------------------------------------------------------------
__global__ void __launch_bounds__(128)
s4_gemm1_kernel(const float* __restrict__ G,
                const float* __restrict__ Xt,
                const float* __restrict__ St,
                float* __restrict__ Yg) {
  const int wid = blockIdx.x * 4 + (threadIdx.x >> 5);
  const int lid = threadIdx.x & 31;
  const int h   = wid >> 3;          // 512 h
  const int j0  = (wid & 7) << 6;    // 8 j-tiles of 64
  const int ml  = lid & 15;
  const int kh  = (lid >> 4) << 1;   // 0 or 2 (K half per lane group)

  const float* gB = G  + h * (KDIM * T);
  const float* xB = Xt + h * (T * JJ);
  const float* sB = St + h * (T * JJ);

  v8f acc[4][4];
#pragma unroll
  for (int mj = 0; mj < 4; ++mj)
#pragma unroll
    for (int nt = 0; nt < 4; ++nt) acc[mj][nt] = {};

  for (int k0 = 0; k0 < KDIM; k0 += 4) {
    const float* src = (k0 < T) ? xB : sB;
    const int kk = (k0 + kh) & (T - 1);
    v2f a[4], bf[4];
#pragma unroll
    for (int m = 0; m < 4; ++m) {
      a[m].x = src[kk * JJ + j0 + 16 * m + ml];
      a[m].y = src[(kk + 1) * JJ + j0 + 16 * m + ml];
    }
#pragma unroll
    for (int n = 0; n < 4; ++n) {
      bf[n].x = gB[(k0 + kh) * T + 16 * n + ml];
      bf[n].y = gB[(k0 + kh + 1) * T + 16 * n + ml];
    }
#pragma unroll
    for (int mj = 0; mj < 4; ++mj)
#pragma unroll
      for (int nt = 0; nt < 4; ++nt)
        acc[mj][nt] = __builtin_amdgcn_wmma_f32_16x16x4_f32(
            false, a[mj], false, bf[nt], (short)0, acc[mj][nt], false, false);
  }

  // GELU + store; p = j*64 + t (= b*L + c*64 + t).
  const int jbase = j0 + ((lid < 16) ? 0 : 8);
#pragma unroll
  for (int mj = 0; mj < 4; ++mj)
#pragma unroll
    for (int nt = 0; nt < 4; ++nt)
#pragma unroll
      for (int v = 0; v < 8; ++v)
        Yg[h * PP + (jbase + 16 * mj + v) * T + 16 * nt + ml] =
            fast_gelu(acc[mj][nt][v]);
}

// ---------------------------------------------------------------------------
// Kernel 5: output projection  Z(1024 x 32768) = W_out(1024x512) @ Yg(512x32768)
// Wave computes 64(o) x 64(p); A-fragment k-pairs are contiguous -> b64 loads.
// Waves in a block share the o-tile (same A data, L0/L2 hits).
// ---------------------------------------------------------------------------
__global__ void __launch_bounds__(128)
s4_gemm2_kernel(const float* __restrict__ W,
                const float* __restrict__ Yg,
                float* __restrict__ Z) {
  const int wid = blockIdx.x * 4 + (threadIdx.x >> 5);
  const int lid = threadIdx.x & 31;
  const int o0  = (wid >> 9) << 6;   // 16 o-tiles of 64
  const int p0  = (wid & 511) << 6;  // 512 p-tiles of 64
  const int ml  = lid & 15;
  const int kh  = (lid >> 4) << 1;

  v8f acc[4][4];
#pragma unroll
  for (int mo = 0; mo < 4; ++mo)
#pragma unroll
    for (int np = 0; np < 4; ++np) acc[mo][np] = {};

  for (int k0 = 0; k0 < HH; k0 += 4) {
    v2f a[4], bf[4];
#pragma unroll
    for (int m = 0; m < 4; ++m)
      a[m] = ld2(&W[(o0 + 16 * m + ml) * HH + k0 + kh]);   // (k,k+1) pair
#pragma unroll
    for (int n = 0; n < 4; ++n) {
      bf[n].x = Yg[(k0 + kh) * PP + p0 + 16 * n + ml];
      bf[n].y = Yg[(k0 + kh + 1) * PP + p0 + 16 * n + ml];
    }
#pragma unroll
    for (int mo = 0; mo < 4; ++mo)
#pragma unroll
      for (int np = 0; np < 4; ++np)
        acc[mo][np] = __builtin_amdgcn_wmma_f32_16x16x4_f32(
            false, a[mo], false, bf[np], (short)0, acc[mo][np], false, false);
  }

  const int obase = o0 + ((lid < 16) ? 0 : 8);
#pragma unroll
  for (int mo = 0; mo < 4; ++mo)
#pragma unroll
    for (int np = 0; np < 4; ++np)
#pragma unroll
      for (int v = 0; v < 8; ++v)
        Z[(obase + 16 * mo + v) * PP + p0 + 16 * np + ml] = acc[mo][np][v];
}

// ---------------------------------------------------------------------------
// Kernel 6: bias + GLU + transpose to (B,L,H): out[p*H + h].
// ---------------------------------------------------------------------------
__global__ void s4_glu_kernel(const float* __restrict__ Z,
                              const float* __restrict__ bo,
                              float* __restrict__ out) {
  const int p0 = blockIdx.x * 32;
  const int h0 = blockIdx.y * 32;
  __shared__ float tile[32][33];
  const int tx = threadIdx.x, ty = threadIdx.y;
#pragma unroll
  for (int i = 0; i < 4; ++i) {
    const int hl = ty + i * 8;
    const int h  = h0 + hl;
    const float z1 = Z[h * PP + p0 + tx] + bo[h];
    const float z2 = Z[(h + HH) * PP + p0 + tx] + bo[h + HH];
    tile[hl][tx] = z1 * fast_sigmoid(z2);
  }
  __syncthreads();
#pragma unroll
  for (int i = 0; i < 4; ++i) {
    const int pl = ty + i * 8;
    out[(p0 + pl) * HH + h0 + tx] = tile[tx][pl];
  }
}

// ---------------------------------------------------------------------------
extern "C" void kernel_launch(void* const* d_in, const int* in_sizes, int n_in,
                              void* d_out, int out_size, void* d_ws, size_t ws_size,
                              hipStream_t stream) {
  (void)in_sizes; (void)n_in; (void)out_size; (void)ws_size;
  const float* x        = (const float*)d_in[0];
  const float* log_dt   = (const float*)d_in[1];
  const float* log_A_re = (const float*)d_in[2];
  const float* A_im     = (const float*)d_in[3];
  const float* C_re     = (const float*)d_in[4];
  const float* C_im     = (const float*)d_in[5];
  const float* D        = (const float*)d_in[6];
  const float* W_out    = (const float*)d_in[7];
  const float* b_out    = (const float*)d_in[8];
  float* out = (float*)d_out;

  float* ws   = (float*)d_ws;
  float* G    = ws;                         // 512*128*64  = 4,194,304 f
  float* Wtab = G    + HH * KDIM * T;       // 512*32*2    =    32,768 f
  float* Xt   = Wtab + HH * NN * 2;         // 512*64*512  = 16,777,216 f
  float* St   = Xt   + HH * T * JJ;         // 512*64*512  = 16,777,216 f
  float* Yg   = St   + HH * T * JJ;         // 512*32768   = 16,777,216 f
  float* Z    = Yg   + HH * PP;             // 1024*32768  = 33,554,432 f
                                            // total ~352 MB

  s4_table_kernel<<<HH, T, 0, stream>>>(log_dt, log_A_re, A_im, C_re, C_im, D, G, Wtab);
  s4_xt_kernel<<<dim3(HH / 32, 2, BB * T), dim3(32, 8), 0, stream>>>(x, Xt);
  s4_scan_kernel<<<(BB * HH) / 4, 128, 0, stream>>>(x, Wtab, St);
  s4_gemm1_kernel<<<(HH * 8) / 4, 128, 0, stream>>>(G, Xt, St, Yg);     // 4096 waves
  s4_gemm2_kernel<<<(16 * 512) / 4, 128, 0, stream>>>(W_out, Yg, Z);    // 8192 waves
  s4_glu_kernel<<<dim3(PP / 32, HH / 32), dim3(32, 8), 0, stream>>>(Z, b_out, out);
}